// MixLoRALayer_22728966931039
// MI455X (gfx1250) — compile-verified
//
#include <hip/hip_runtime.h>

#define B_ 8
#define S_ 2048
#define IN_ 4096
#define OUT_ 4096
#define R_ 16
#define E_ 64

typedef __attribute__((ext_vector_type(2))) float v2f;
typedef __attribute__((ext_vector_type(8))) float v8f;

// ---------- routing: g_A scores (B x E GEMV) ----------
__global__ void k_scoresA(const float* __restrict__ qs, const float* __restrict__ W,
                          const float* __restrict__ bias, float* __restrict__ g) {
  const int e = blockIdx.x, b = blockIdx.y, t = threadIdx.x;
  __shared__ float red[64];
  const float* q = qs + (size_t)b * IN_;
  const float* w = W + (size_t)e * IN_;
  float s = 0.f;
  for (int i = t; i < IN_; i += 64) s += q[i] * w[i];
  red[t] = s; __syncthreads();
  for (int off = 32; off > 0; off >>= 1) { if (t < off) red[t] += red[t + off]; __syncthreads(); }
  if (t == 0) g[b * E_ + e] = red[0] + bias[e];
}

// ---------- top-16 of 64, descending, stable (lowest index wins ties) ----------
__global__ void k_topk(const float* __restrict__ g, int* __restrict__ idx) {
  const int b = threadIdx.x;
  if (b >= B_) return;
  unsigned long long used = 0ull;
  for (int r = 0; r < R_; ++r) {
    float best = -__builtin_inff(); int bi = 0;
    for (int e = 0; e < E_; ++e) {
      if ((used >> e) & 1ull) continue;
      float v = g[b * E_ + e];
      if (v > best) { best = v; bi = e; }
    }
    used |= (1ull << bi);
    idx[b * R_ + r] = bi;
  }
}

// ---------- g_B scores: GEMV + einsum('bri,rie->be') term ----------
__global__ void k_scoresB(const float* __restrict__ qs, const float* __restrict__ W,
                          const float* __restrict__ bias, const float* __restrict__ A_pool,
                          const float* __restrict__ cfs_W, const int* __restrict__ idxA,
                          float* __restrict__ g) {
  const int e = blockIdx.x, b = blockIdx.y, t = threadIdx.x;
  __shared__ float red[256];
  const float* q = qs + (size_t)b * IN_;
  const float* w = W + (size_t)e * IN_;
  float s = 0.f;
  for (int i = t; i < IN_; i += 256) s += q[i] * w[i];
  for (int j = t; j < R_ * IN_; j += 256) {
    const int r = j >> 12;                      // IN_ == 4096
    const int i = j & (IN_ - 1);
    const int ex = idxA[b * R_ + r];
    s += A_pool[((size_t)ex * R_ + r) * IN_ + i] * cfs_W[(size_t)j * E_ + e];
  }
  red[t] = s; __syncthreads();
  for (int off = 128; off > 0; off >>= 1) { if (t < off) red[t] += red[t + off]; __syncthreads(); }
  if (t == 0) g[b * E_ + e] = red[0] + bias[e];
}

// ---------- gather lora_A[b][r][i] and lora_B packed as [b][o][k] ----------
__global__ void k_gather(const float* __restrict__ A_pool, const float* __restrict__ B_pool,
                         const int* __restrict__ idxA, const int* __restrict__ idxB,
                         float* __restrict__ loraA, float* __restrict__ loraBp) {
  const int t = blockIdx.x * blockDim.x + threadIdx.x;
  if (t < B_ * R_ * IN_) {
    const int b = t / (R_ * IN_);
    const int r = (t / IN_) % R_;
    const int i = t % IN_;
    loraA[t] = A_pool[((size_t)idxA[b * R_ + r] * R_ + r) * IN_ + i];
  }
  if (t < B_ * OUT_ * R_) {
    const int b = t / (OUT_ * R_);
    const int o = (t / R_) % OUT_;
    const int k = t % R_;
    // loraBp[b][o][k] = lora_B[b][k][o] = B_pool[idxB[b][k]][o][k]
    loraBp[t] = B_pool[((size_t)idxB[b * R_ + k] * OUT_ + o) * R_ + k];
  }
}

// ---------- fused main kernel ----------
// One block per 16-row M-tile. Phase 1: 8 waves split K (512 each), partial 16x16
// accumulators reduced through LDS. Phase 2: 8 waves split OUT (512 cols each).
__global__ void __launch_bounds__(256) k_main(const float* __restrict__ x,
                                              const float* __restrict__ loraA,
                                              const float* __restrict__ loraBp,
                                              float* __restrict__ out) {
  __shared__ __align__(16) float part[8][16][16];  // per-wave partial after_A [m][r]
  __shared__ __align__(16) float redt[16][16];     // reduced after_A tile
  const int lane = threadIdx.x & 31;
  const int wave = threadIdx.x >> 5;
  const int half = lane >> 4;            // K sub-group select per ISA A/B layout
  const int l15  = lane & 15;
  const int b    = blockIdx.y;
  const int m0   = blockIdx.x * 16;
  const int t    = threadIdx.x;

  // ---- phase 1: partial after_A over this wave's K chunk (512) ----
  v8f acc = {};
  const int kbase = wave * (IN_ / 8);
  const float* xrow = x     + ((size_t)b * S_ + m0 + l15) * IN_ + kbase + 2 * half;
  const float* arow = loraA + ((size_t)b * R_ + l15) * IN_ + kbase + 2 * half;
  for (int k0 = 0; k0 < IN_ / 8; k0 += 8) {
    v2f a0 = __builtin_nontemporal_load((const v2f*)(xrow + k0));
    v2f b0 = *(const v2f*)(arow + k0);
    acc = __builtin_amdgcn_wmma_f32_16x16x4_f32(false, a0, false, b0, (short)0, acc, false, false);
    v2f a1 = __builtin_nontemporal_load((const v2f*)(xrow + k0 + 4));
    v2f b1 = *(const v2f*)(arow + k0 + 4);
    acc = __builtin_amdgcn_wmma_f32_16x16x4_f32(false, a1, false, b1, (short)0, acc, false, false);
  }

  // ---- spill partials (C layout: VGPR j -> row j + 8*half, col l15) ----
#pragma unroll
  for (int j = 0; j < 8; ++j) part[wave][j + 8 * half][l15] = acc[j];
  __syncthreads();

  // ---- cross-wave reduction: 256 threads, one element each ----
  {
    const int m = t >> 4, r = t & 15;
    float s = 0.f;
#pragma unroll
    for (int w = 0; w < 8; ++w) s += part[w][m][r];
    redt[m][r] = s;
  }
  __syncthreads();

  // ---- phase 2 A-fragments from reduced tile (M=l15, K=4kk+2half+{0,1}) ----
  v2f afrag[4];
#pragma unroll
  for (int kk = 0; kk < 4; ++kk)
    afrag[kk] = *(const v2f*)&redt[l15][4 * kk + 2 * half];

  // ---- phase 2: this wave's 512-column chunk of OUT ----
  const float* bp   = loraBp + ((size_t)b * OUT_) * R_;       // [o][k], k contiguous
  float*       orow = out + ((size_t)b * S_ + m0) * OUT_;
  const int nbase = wave * (OUT_ / 8);
  for (int n0 = nbase; n0 < nbase + OUT_ / 8; n0 += 16) {
    v8f c = {};
#pragma unroll
    for (int kk = 0; kk < 4; ++kk) {
      v2f bf = *(const v2f*)(bp + (size_t)(n0 + l15) * R_ + 4 * kk + 2 * half);
      c = __builtin_amdgcn_wmma_f32_16x16x4_f32(false, afrag[kk], false, bf, (short)0, c, false, false);
    }
#pragma unroll
    for (int j = 0; j < 8; ++j)
      __builtin_nontemporal_store(c[j], &orow[(size_t)(j + 8 * half) * OUT_ + n0 + l15]);
  }
}

extern "C" void kernel_launch(void* const* d_in, const int* in_sizes, int n_in,
                              void* d_out, int out_size, void* d_ws, size_t ws_size,
                              hipStream_t stream) {
  const float* x      = (const float*)d_in[0];
  const float* qs     = (const float*)d_in[1];
  const float* A_pool = (const float*)d_in[2];
  const float* B_pool = (const float*)d_in[3];
  const float* W_rA   = (const float*)d_in[4];
  const float* b_rA   = (const float*)d_in[5];
  const float* W_rB   = (const float*)d_in[6];
  const float* b_rB   = (const float*)d_in[7];
  const float* cfs_W  = (const float*)d_in[8];
  float* out = (float*)d_out;

  char* ws = (char*)d_ws;
  float* gA     = (float*)(ws + 0);
  float* gB     = (float*)(ws + 4096);
  int*   idxA   = (int*)(ws + 8192);
  int*   idxB   = (int*)(ws + 12288);
  float* loraA  = (float*)(ws + 16384);
  float* loraBp = (float*)(ws + 16384 + (size_t)B_ * R_ * IN_ * sizeof(float));

  k_scoresA<<<dim3(E_, B_), 64, 0, stream>>>(qs, W_rA, b_rA, gA);
  k_topk<<<1, 32, 0, stream>>>(gA, idxA);
  k_scoresB<<<dim3(E_, B_), 256, 0, stream>>>(qs, W_rB, b_rB, A_pool, cfs_W, idxA, gB);
  k_topk<<<1, 32, 0, stream>>>(gB, idxB);
  k_gather<<<(B_ * OUT_ * R_ + 255) / 256, 256, 0, stream>>>(A_pool, B_pool, idxA, idxB, loraA, loraBp);
  k_main<<<dim3(S_ / 16, B_), 256, 0, stream>>>(x, loraA, loraBp, out);
}